// KinematicSfxnNetwork_88158498718335
// MI455X (gfx1250) — compile-verified
//
#include <hip/hip_runtime.h>
#include <hip/hip_bf16.h>
#include <math.h>

#define SIGMA2 3.61f      // 1.9*1.9
#define EPS_LJ 0.2f
#define R2_SOFT 0.25f
#define SEGS 8            // j-segments per row-pair (occupancy)

typedef __attribute__((ext_vector_type(2))) float v2f;
typedef __attribute__((ext_vector_type(8))) float v8f;

// ---------------- affine 3x4 transform helpers (bottom row [0,0,0,1]) -------
struct Aff { float m[12]; };   // row-major 3x4: [R | t]

__device__ __forceinline__ Aff aff_identity() {
    Aff c;
#pragma unroll
    for (int i = 0; i < 12; ++i) c.m[i] = 0.0f;
    c.m[0] = c.m[5] = c.m[10] = 1.0f;
    return c;
}

// C = A @ B (4x4 homogeneous product restricted to affine rows)
__device__ __forceinline__ Aff aff_compose(const Aff& a, const Aff& b) {
    Aff c;
#pragma unroll
    for (int r = 0; r < 3; ++r) {
        float a0 = a.m[r*4+0], a1 = a.m[r*4+1], a2 = a.m[r*4+2];
#pragma unroll
        for (int col = 0; col < 3; ++col)
            c.m[r*4+col] = a0*b.m[0*4+col] + a1*b.m[1*4+col] + a2*b.m[2*4+col];
        c.m[r*4+3] = a0*b.m[3] + a1*b.m[7] + a2*b.m[11] + a.m[r*4+3];
    }
    return c;
}

// local HT from (phi, theta, d) bond dofs
__device__ __forceinline__ Aff local_ht(const float* __restrict__ dof) {
    float phi = dof[0], theta = dof[1];
    float d = 1.5f + 0.25f * tanhf(dof[2]);
    float sp, cp, st, ct;
    sincosf(phi, &sp, &cp);
    sincosf(theta, &st, &ct);
    Aff h;
    h.m[0] = cp*ct;  h.m[1] = -sp;   h.m[2]  = cp*st;
    h.m[4] = sp*ct;  h.m[5] =  cp;   h.m[6]  = sp*st;
    h.m[8] = -st;    h.m[9] = 0.0f;  h.m[10] = ct;
    h.m[3]  = d*h.m[0];
    h.m[7]  = d*h.m[4];
    h.m[11] = d*h.m[8];
    return h;
}

// ------------- kernel 1: per-pose forward-kinematics scan + scatter ---------
__global__ __launch_bounds__(256)
void fk_scan_kernel(const float* __restrict__ dofs,
                    const int* __restrict__ kid,
                    float* __restrict__ coords,   // [P*A*3], pre-filled w/ pose coords
                    int A) {
    __shared__ Aff buf0[256];
    __shared__ Aff buf1[256];
    const int p   = blockIdx.x;
    const int tid = threadIdx.x;
    const int CH  = A / 256;                 // 16 atoms per thread
    const int a0  = tid * CH;
    const float* dbase = dofs + ((size_t)p * A) * 9;

    // pass 1: serial product of this thread's chunk
    Aff acc = local_ht(dbase + (size_t)a0 * 9);
    for (int k = 1; k < CH; ++k)
        acc = aff_compose(acc, local_ht(dbase + (size_t)(a0 + k) * 9));
    buf0[tid] = acc;
    __syncthreads();

    // Hillis-Steele inclusive scan over 256 chunk products (order-sensitive)
    Aff* src = buf0; Aff* dst = buf1;
    for (int step = 1; step < 256; step <<= 1) {
        if (tid >= step) dst[tid] = aff_compose(src[tid - step], src[tid]);
        else             dst[tid] = src[tid];
        __syncthreads();
        Aff* t = src; src = dst; dst = t;
    }
    Aff pre = (tid > 0) ? src[tid - 1] : aff_identity();

    // pass 2: rebuild per-atom cumulative transforms, scatter translations
    acc = pre;
    for (int k = 0; k < CH; ++k) {
        acc = aff_compose(acc, local_ht(dbase + (size_t)(a0 + k) * 9));
        int idx = kid[(size_t)p * A + a0 + k];
        coords[(size_t)idx * 3 + 0] = acc.m[3];
        coords[(size_t)idx * 3 + 1] = acc.m[7];
        coords[(size_t)idx * 3 + 2] = acc.m[11];
    }
}

// ------------- kernel 2: pad to xyz0 (float4) + squared norms ---------------
// coords4 layout makes WMMA fragment loads a single unconditional b64:
//   lo lane -> (x,y) = K0,K1 ; hi lane (base+8B) -> (z,0) = K2,K3
__global__ __launch_bounds__(256)
void pack_sq_kernel(const float* __restrict__ coords,
                    float* __restrict__ coords4,
                    float* __restrict__ sq, int n) {
    int i = blockIdx.x * blockDim.x + threadIdx.x;
    if (i < n) {
        float x = coords[(size_t)i*3+0], y = coords[(size_t)i*3+1], z = coords[(size_t)i*3+2];
        float4 c4 = make_float4(x, y, z, 0.0f);
        *(float4*)(coords4 + (size_t)i * 4) = c4;
        sq[i] = x*x + y*y + z*z;
    }
}

// ------------- kernel 3: LJ via V_WMMA_F32_16X16X4_F32 over tile rows ------
// Balanced triangle partition: wave handles row-pair (r, NT-1-r) = NT+1 tiles,
// split into SEGS segments. Tile mapping forced into SGPRs (readfirstlane) so
// all control flow is scalar and EXEC stays all-1s around the WMMA.
__global__ __launch_bounds__(256)
void lj_wmma_kernel(const float* __restrict__ coords4,
                    const float* __restrict__ sq,
                    float* __restrict__ partials,
                    int A, int NT, int totalWaves) {
    const int gw = __builtin_amdgcn_readfirstlane(
        (int)((blockIdx.x * blockDim.x + threadIdx.x) >> 5));
    const int lane = threadIdx.x & 31;
    if (gw >= totalWaves) return;             // scalar branch

    const int perPose = (NT / 2) * SEGS;
    const int p   = gw / perPose;
    const int w   = gw - p * perPose;
    const int rp  = w / SEGS;                 // row-pair index
    const int seg = w - rp * SEGS;
    const int r1  = rp, r2 = NT - 1 - rp;
    const int U   = NT + 1;                   // tiles in this row-pair
    const int u0  = (seg * U) / SEGS;
    const int u1  = ((seg + 1) * U) / SEGS;
    const int row1len = NT - r1;              // tiles belonging to row r1

    const float* cp4 = coords4 + (size_t)p * A * 4;
    const float* sp2 = sq + (size_t)p * A;
    const int  l  = lane & 15;
    const bool hi = lane >= 16;
    const int  mbase = hi ? 8 : 0;
    const int  fragoff = l * 4 + (hi ? 2 : 0);   // float offset within a 16-atom tile

    float acc = 0.0f;
    int curRow = -1;
    v2f a = {0.0f, 0.0f};
    float t1[8];
#pragma unroll
    for (int v = 0; v < 8; ++v) t1[v] = 0.0f;

    for (int u = u0; u < u1; ++u) {
        int row, j;
        if (u < row1len) { row = r1; j = r1 + u; }   // scalar select
        else             { row = r2; j = r2 + (u - row1len); }

        if (row != curRow) {                  // scalar; at most twice per wave
            curRow = row;
            const int i0 = row * 16;
            a = *(const v2f*)(cp4 + (size_t)i0 * 4 + fragoff);   // one b64 load
#pragma unroll
            for (int v = 0; v < 8; ++v)
                t1[v] = sp2[i0 + mbase + v] + R2_SOFT;
        }

        const int j0 = j * 16;
        v2f b = *(const v2f*)(cp4 + (size_t)j0 * 4 + fragoff);   // one b64 load

        v8f c = {};
        v8f d = __builtin_amdgcn_wmma_f32_16x16x4_f32(false, a, false, b,
                                                      (short)0, c, false, false);
        const float sqj = sp2[j0 + l];

        if (j == curRow) {                    // scalar branch: diagonal tile
#pragma unroll
            for (int v = 0; v < 8; ++v) {
                float r2v = t1[v] + (sqj - 2.0f * d[v]);
                float s   = SIGMA2 * __builtin_amdgcn_rcpf(r2v);
                float i6  = s * s * s;
                float e   = i6 * i6 - i6;
                acc += ((mbase + v) < l) ? e : 0.0f;   // strict upper mask
            }
        } else {                              // off-diagonal: all 256 pairs count
#pragma unroll
            for (int v = 0; v < 8; ++v) {
                float r2v = t1[v] + (sqj - 2.0f * d[v]);
                float s   = SIGMA2 * __builtin_amdgcn_rcpf(r2v);
                float i6  = s * s * s;
                acc += i6 * i6 - i6;
            }
        }
    }
    acc *= 4.0f * EPS_LJ;                     // fold constant out of inner loop

    // wave32 reduction, one partial per wave
#pragma unroll
    for (int off = 16; off > 0; off >>= 1) acc += __shfl_xor(acc, off, 32);
    if (lane == 0) partials[gw] = acc;
}

// ------------- kernel 4: deterministic per-pose reduction ------------------
__global__ __launch_bounds__(256)
void reduce_kernel(const float* __restrict__ partials, float* __restrict__ out,
                   int perPose) {
    __shared__ float s[256];
    const int p = blockIdx.x;
    float acc = 0.0f;
    for (int i = threadIdx.x; i < perPose; i += 256)
        acc += partials[(size_t)p * perPose + i];
    s[threadIdx.x] = acc;
    __syncthreads();
#pragma unroll
    for (int off = 128; off > 0; off >>= 1) {
        if (threadIdx.x < off) s[threadIdx.x] += s[threadIdx.x + off];
        __syncthreads();
    }
    if (threadIdx.x == 0) out[p] = s[0];
}

// ---------------------------------------------------------------------------
extern "C" void kernel_launch(void* const* d_in, const int* in_sizes, int n_in,
                              void* d_out, int out_size, void* d_ws, size_t ws_size,
                              hipStream_t stream) {
    const float* dofs = (const float*)d_in[0];   // [P*A, 9]
    const float* pose = (const float*)d_in[1];   // [P, A, 3]
    const int*   kid  = (const int*)d_in[2];     // [P*A]
    float* out = (float*)d_out;                  // [P]

    const int P  = out_size;               // 4
    const int PA = in_sizes[1] / 3;        // P*A
    const int A  = PA / P;                 // 4096
    const int NT = A / 16;                 // 256 tiles per dim
    const int perPose = (NT / 2) * SEGS;   // waves (partials) per pose
    const int totalWaves = P * perPose;    // 4096

    float* ws_coords   = (float*)d_ws;              // PA*3 floats (scatter target)
    float* ws_coords4  = ws_coords + (size_t)PA*3;  // PA*4 floats (xyz0 padded)
    float* ws_sq       = ws_coords4 + (size_t)PA*4; // PA floats
    float* ws_partials = ws_sq + PA;                // totalWaves floats

    // scatter base = pose_stack_coords
    hipMemcpyAsync(ws_coords, pose, (size_t)PA * 3 * sizeof(float),
                   hipMemcpyDeviceToDevice, stream);

    fk_scan_kernel<<<P, 256, 0, stream>>>(dofs, kid, ws_coords, A);
    pack_sq_kernel<<<(PA + 255) / 256, 256, 0, stream>>>(ws_coords, ws_coords4,
                                                         ws_sq, PA);

    const int wavesPerBlock = 256 / 32;
    const int nb = (totalWaves + wavesPerBlock - 1) / wavesPerBlock;
    lj_wmma_kernel<<<nb, 256, 0, stream>>>(ws_coords4, ws_sq, ws_partials,
                                           A, NT, totalWaves);
    reduce_kernel<<<P, 256, 0, stream>>>(ws_partials, out, perPose);
}